// WarpTranspose_44848048504965
// MI455X (gfx1250) — compile-verified
//
#include <hip/hip_runtime.h>

// Problem constants from the reference: B=8, H=512, W=512, C=16 (all power-of-2).
constexpr int Bn = 8;
constexpr int Hn = 512;
constexpr int Wn = 512;
constexpr int Cn = 16;

typedef float v4f __attribute__((ext_vector_type(4)));

// ---------------------------------------------------------------------------
// Pass 1: zero the output (harness poisons d_out with 0xAA before timing).
// 128-bit stores, fully coalesced.
// ---------------------------------------------------------------------------
__global__ __launch_bounds__(256) void WarpT_zero_kernel(v4f* __restrict__ out, int n4) {
  int i = blockIdx.x * 256 + threadIdx.x;
  if (i < n4) {
    v4f z = {0.f, 0.f, 0.f, 0.f};
    out[i] = z;
  }
}

// ---------------------------------------------------------------------------
// Pass 2: flow-guided bilinear scatter-add.
// Mapping: 4 threads per pixel; each thread owns 4 contiguous channels.
//   tid -> pix = tid>>2, channel group cg = (tid&3)*4.
// Consecutive lanes cover the 64B channel row of a pixel -> b128 loads are
// perfectly coalesced. Scatter uses non-returning global_atomic_add_f32
// (resolved at L2; STOREcnt fire-and-forget).
// grad_out is streamed exactly once -> non-temporal load hint so the 192MB L2
// stays available for the 128MB scatter-target working set.
// ---------------------------------------------------------------------------
__device__ __forceinline__ void scatter4(float* __restrict__ p, float w, v4f g) {
  atomicAdd(p + 0, w * g.x);
  atomicAdd(p + 1, w * g.y);
  atomicAdd(p + 2, w * g.z);
  atomicAdd(p + 3, w * g.w);
}

__global__ __launch_bounds__(256) void WarpT_scatter_kernel(
    const float* __restrict__ grad_out,   // [B,H,W,C]
    const float2* __restrict__ u,         // [B,H,W,2]  (.x = dx, .y = dy)
    float* __restrict__ grad_x)           // [B,H,W,C]
{
  const int tid = blockIdx.x * 256 + threadIdx.x;
  const int pix = tid >> 2;          // flat (b,y,x) pixel index
  const int cg  = (tid & 3) << 2;    // channel group: 0,4,8,12

  const int x     = pix & (Wn - 1);
  const int y     = (pix >> 9) & (Hn - 1);
  const int bBase = pix & ~(Hn * Wn - 1);   // b * H * W

  // Streamed once: NT hint keeps these lines out of the way in L2.
  const v4f g = __builtin_nontemporal_load(
      (const v4f*)(grad_out + (size_t)pix * Cn + cg));
  const float2 uv = u[pix];  // 4 lanes/pixel share one cacheline; L0 absorbs it

  const float gx  = (float)x + uv.x;
  const float gy  = (float)y + uv.y;
  const float x0f = floorf(gx);
  const float y0f = floorf(gy);
  const float wx  = gx - x0f;
  const float wy  = gy - y0f;
  const int x0 = (int)x0f, y0 = (int)y0f;
  const int x1 = x0 + 1,   y1 = y0 + 1;

  const bool vx0 = (unsigned)x0 < (unsigned)Wn;   // mode='zeros' boundary:
  const bool vx1 = (unsigned)x1 < (unsigned)Wn;   // out-of-range corners are
  const bool vy0 = (unsigned)y0 < (unsigned)Hn;   // simply dropped (w -> 0)
  const bool vy1 = (unsigned)y1 < (unsigned)Hn;

  float* const outb = grad_x + (size_t)bBase * Cn + cg;
  if (vy0) {
    float* row = outb + (size_t)y0 * (Wn * Cn);
    if (vx0) scatter4(row + x0 * Cn, (1.f - wy) * (1.f - wx), g);
    if (vx1) scatter4(row + x1 * Cn, (1.f - wy) * wx,         g);
  }
  if (vy1) {
    float* row = outb + (size_t)y1 * (Wn * Cn);
    if (vx0) scatter4(row + x0 * Cn, wy * (1.f - wx), g);
    if (vx1) scatter4(row + x1 * Cn, wy * wx,         g);
  }
}

// ---------------------------------------------------------------------------
// Harness entry point. Inputs: d_in[0]=grad_out f32, d_in[1]=u f32,
// d_in[2]=x (shape/dtype only, unused). Output: grad_x f32, out_size elems.
// ---------------------------------------------------------------------------
extern "C" void kernel_launch(void* const* d_in, const int* in_sizes, int n_in,
                              void* d_out, int out_size, void* d_ws, size_t ws_size,
                              hipStream_t stream) {
  (void)in_sizes; (void)n_in; (void)d_ws; (void)ws_size;

  const float*  grad_out = (const float*)d_in[0];
  const float2* u        = (const float2*)d_in[1];
  float*        grad_x   = (float*)d_out;

  // Zero-init output (out_size = 8*512*512*16 = 33,554,432 floats).
  const int n4 = out_size >> 2;                    // float4 count
  const int zb = (n4 + 255) / 256;
  WarpT_zero_kernel<<<zb, 256, 0, stream>>>((v4f*)grad_x, n4);

  // Scatter pass: 4 threads per pixel.
  const int nthreads = Bn * Hn * Wn * 4;           // 8,388,608
  WarpT_scatter_kernel<<<nthreads / 256, 256, 0, stream>>>(grad_out, u, grad_x);
}